// FlowNet3D_49709951484218
// MI455X (gfx1250) — compile-verified
//
#include <hip/hip_runtime.h>
#include <vector>
#include <initializer_list>

#define Bb 8
#define Nn 4096

typedef float v2f __attribute__((ext_vector_type(2)));
typedef float v8f __attribute__((ext_vector_type(8)));

__device__ __forceinline__ float sqr(float x){ return x*x; }

// ---------------- layout conversion: [B,3,n] -> channel-major [3][B*n] ----------
__global__ void k_chanmajor(const float* __restrict__ in, float* __restrict__ out, int n){
  int t = blockIdx.x*blockDim.x + threadIdx.x;
  int total = Bb*3*n;
  if (t >= total) return;
  int i = t % n; int c = (t/n)%3; int b = t/(3*n);
  out[c*(Bb*n) + b*n + i] = in[t];
}

// ---------------- furthest point sampling: one block per batch ------------------
__global__ void k_fps(const float* __restrict__ xyz, int n, int S, int* __restrict__ out){
  int b = blockIdx.x;
  const float* X = xyz + b*n;
  const float* Y = xyz + Bb*n + b*n;
  const float* Z = xyz + 2*Bb*n + b*n;
  extern __shared__ float dist[];              // n floats
  __shared__ float rmax[256]; __shared__ int rarg[256];
  for (int i = threadIdx.x; i < n; i += blockDim.x) dist[i] = 1e10f;
  __syncthreads();
  int far = 0;
  for (int s = 0; s < S; ++s){
    if (threadIdx.x == 0) out[b*S + s] = far;
    float cx = X[far], cy = Y[far], cz = Z[far];
    float best = -1.f; int bi = 0;
    for (int i = threadIdx.x; i < n; i += blockDim.x){
      float d = sqr(X[i]-cx)+sqr(Y[i]-cy)+sqr(Z[i]-cz);
      float nd = fminf(dist[i], d);
      dist[i] = nd;
      if (nd > best){ best = nd; bi = i; }
    }
    rmax[threadIdx.x] = best; rarg[threadIdx.x] = bi;
    __syncthreads();
    for (int off = blockDim.x/2; off > 0; off >>= 1){
      if (threadIdx.x < off){
        float o = rmax[threadIdx.x+off]; int oi = rarg[threadIdx.x+off];
        if (o > rmax[threadIdx.x] || (o == rmax[threadIdx.x] && oi < rarg[threadIdx.x])){
          rmax[threadIdx.x] = o; rarg[threadIdx.x] = oi;
        }
      }
      __syncthreads();
    }
    far = rarg[0];
    __syncthreads();
  }
}

// ---------------- gather sampled coordinates -----------------------------------
__global__ void k_gather3(const float* __restrict__ src, int n, const int* __restrict__ idx,
                          int S, float* __restrict__ dst){
  int t = blockIdx.x*blockDim.x + threadIdx.x;
  if (t >= Bb*S) return;
  int b = t / S;
  int j = idx[t];
  for (int c = 0; c < 3; ++c) dst[c*(Bb*S)+t] = src[c*(Bb*n)+b*n+j];
}

// ---------------- ball query: first K in-range indices (ascending) --------------
__global__ void k_ball(const float* __restrict__ xyz, int n, const float* __restrict__ q,
                       int S, float r2, int K, int* __restrict__ idx){
  int t = blockIdx.x*blockDim.x + threadIdx.x;
  if (t >= Bb*S) return;
  int b = t / S;
  float qx = q[t], qy = q[Bb*S+t], qz = q[2*Bb*S+t];
  const float* X = xyz + b*n; const float* Y = xyz + Bb*n + b*n; const float* Z = xyz + 2*Bb*n + b*n;
  int* o = idx + (long)t*K;
  int cnt = 0, first = 0; bool has = false;
  for (int i = 0; i < n && cnt < K; ++i){
    float d = sqr(X[i]-qx)+sqr(Y[i]-qy)+sqr(Z[i]-qz);
    if (d <= r2){ if (!has){ first = i; has = true; } o[cnt++] = i; }
  }
  for (; cnt < K; ++cnt) o[cnt] = has ? first : 0;
}

// ---------------- SA grouping: X = concat(grouped_xyz - center, grouped_pts) ----
__global__ void k_group_sa(const float* __restrict__ xyz, const float* __restrict__ pts,
                           int n, int D, const float* __restrict__ newxyz,
                           const int* __restrict__ idx, int S, int K, float* __restrict__ Xo){
  long cols = (long)Bb*S*K;
  long t = (long)blockIdx.x*blockDim.x + threadIdx.x;
  if (t >= cols) return;
  long sBS = t / K;
  int b = (int)(sBS / S);
  int j = idx[t];
  for (int c = 0; c < 3; ++c)
    Xo[(long)c*cols + t] = xyz[c*(Bb*n)+b*n+j] - newxyz[c*(Bb*S)+sBS];
  for (int c = 0; c < D; ++c)
    Xo[(long)(3+c)*cols + t] = pts[(long)c*(Bb*n)+b*n+j];
}

// ---------------- WMMA fp32 GEMM: Y[Co,cols] = W[Co,Ci] * X[Ci,cols] (+bias) ----
// 2 M-tiles x 4 N-tiles per wave; K-loop step 4 via V_WMMA_F32_16X16X4_F32.
// Hot loop is branch-free: row/col indices are clamped up front (cols is always
// a multiple of 16 in this pipeline), only stores are masked at the edges.
// A (16x4 fp32): lanes 0-15 hold K={k0,k0+1}, lanes 16-31 K={k0+2,k0+3} (ISA 7.12.2)
__global__ void k_gemm_wmma(const float* __restrict__ W, const float* __restrict__ Xm,
                            const float* __restrict__ bias, float* __restrict__ Y,
                            int Co, int Ci, long cols){
  constexpr int MT = 2, NT = 4;
  int lane = threadIdx.x & 31;
  long wid = (long)blockIdx.x * (blockDim.x >> 5) + (threadIdx.x >> 5);
  int  mTiles  = (Co + 15) >> 4;
  int  mGroups = (mTiles + MT - 1) / MT;
  long nGroups = (cols + 16*NT - 1) / (16*NT);
  if (wid >= (long)mGroups * nGroups) return;
  int  mg = (int)(wid % mGroups);
  long ng = wid / mGroups;
  int  m0 = mg * (16*MT);
  long n0 = ng * (16*NT);
  int half = lane >> 4, lid = lane & 15;
  int kbase = 2*half;

  // clamped row/col bases so the hot loop needs no guards
  long wrow[MT];
#pragma unroll
  for (int i = 0; i < MT; ++i){
    int mr = m0 + 16*i + lid; if (mr > Co-1) mr = Co-1;
    wrow[i] = (long)mr * Ci;
  }
  long ncol[NT];
#pragma unroll
  for (int j = 0; j < NT; ++j){
    long nc = n0 + 16*j + lid; if (nc > cols-1) nc = cols-1;
    ncol[j] = nc;
  }

  const v8f vzero = {0.f,0.f,0.f,0.f,0.f,0.f,0.f,0.f};
  v8f acc[MT][NT];
#pragma unroll
  for (int i = 0; i < MT; ++i)
#pragma unroll
    for (int j = 0; j < NT; ++j) acc[i][j] = vzero;

  int CiA = Ci & ~3;
  for (int k0 = 0; k0 < CiA; k0 += 4){
    v2f a[MT], b[NT];
#pragma unroll
    for (int i = 0; i < MT; ++i)
#pragma unroll
      for (int u = 0; u < 2; ++u)
        a[i][u] = W[wrow[i] + k0 + kbase + u];
#pragma unroll
    for (int j = 0; j < NT; ++j)
#pragma unroll
      for (int u = 0; u < 2; ++u)
        b[j][u] = Xm[(long)(k0 + kbase + u)*cols + ncol[j]];
    if (k0 + 4 < CiA) __builtin_prefetch(&Xm[(long)(k0 + 4 + kbase)*cols + ncol[0]], 0, 1);
#pragma unroll
    for (int i = 0; i < MT; ++i)
#pragma unroll
      for (int j = 0; j < NT; ++j)
        acc[i][j] = __builtin_amdgcn_wmma_f32_16x16x4_f32(false, a[i], false, b[j],
                                                          (short)0, acc[i][j], false, false);
  }
  if (CiA < Ci){
    // single branch-free tail step: clamp k, zero the A fragment past Ci
    v2f a[MT], b[NT];
#pragma unroll
    for (int i = 0; i < MT; ++i)
#pragma unroll
      for (int u = 0; u < 2; ++u){
        int kk = CiA + kbase + u;
        int kc = kk < Ci ? kk : Ci-1;
        float av = W[wrow[i] + kc];
        a[i][u] = kk < Ci ? av : 0.f;
      }
#pragma unroll
    for (int j = 0; j < NT; ++j)
#pragma unroll
      for (int u = 0; u < 2; ++u){
        int kk = CiA + kbase + u;
        int kc = kk < Ci ? kk : Ci-1;
        b[j][u] = Xm[(long)kc*cols + ncol[j]];
      }
#pragma unroll
    for (int i = 0; i < MT; ++i)
#pragma unroll
      for (int j = 0; j < NT; ++j)
        acc[i][j] = __builtin_amdgcn_wmma_f32_16x16x4_f32(false, a[i], false, b[j],
                                                          (short)0, acc[i][j], false, false);
  }

  // store: fast path for full M-tiles (uniform), masked path only at the ragged edge
#pragma unroll
  for (int i = 0; i < MT; ++i){
    bool fullM = (m0 + 16*i + 16) <= Co;
#pragma unroll
    for (int j = 0; j < NT; ++j){
      if (n0 + 16*j < cols){                       // wave-uniform tile validity
        long ncolS = n0 + 16*j + lid;
        if (fullM){
#pragma unroll
          for (int r = 0; r < 8; ++r){
            int mm = m0 + 16*i + r + 8*half;
            float v = acc[i][j][r];
            if (bias) v += bias[mm];
            Y[(long)mm*cols + ncolS] = v;
          }
        } else {
#pragma unroll
          for (int r = 0; r < 8; ++r){
            int mm = m0 + 16*i + r + 8*half;
            if (mm < Co){
              float v = acc[i][j][r];
              if (bias) v += bias[mm];
              Y[(long)mm*cols + ncolS] = v;
            }
          }
        }
      }
    }
  }
}

// ---------------- BN: deterministic per-channel reduction, then apply -----------
__global__ void k_bn_reduce(const float* __restrict__ Y, long cols, float* __restrict__ stats){
  int c = blockIdx.x;
  const float* row = Y + (long)c*cols;
  float s = 0.f, s2 = 0.f;
  for (long i = threadIdx.x; i < cols; i += blockDim.x){ float v = row[i]; s += v; s2 += v*v; }
  __shared__ float rs[256], rs2[256];
  rs[threadIdx.x] = s; rs2[threadIdx.x] = s2; __syncthreads();
  for (int off = blockDim.x/2; off > 0; off >>= 1){
    if (threadIdx.x < off){ rs[threadIdx.x] += rs[threadIdx.x+off]; rs2[threadIdx.x] += rs2[threadIdx.x+off]; }
    __syncthreads();
  }
  if (threadIdx.x == 0){
    float mean = rs[0]/(float)cols;
    stats[2*c] = mean; stats[2*c+1] = rs2[0]/(float)cols - mean*mean;
  }
}

__global__ void k_bn_apply(float* __restrict__ Y, long cols, int Co,
                           const float* __restrict__ stats, const float* __restrict__ g,
                           const float* __restrict__ bb, int relu){
  long t = (long)blockIdx.x*blockDim.x + threadIdx.x;
  if (t >= (long)Co*cols) return;
  int c = (int)(t / cols);
  float v = (Y[t] - stats[2*c]) * rsqrtf(stats[2*c+1] + 1e-5f) * g[c] + bb[c];
  if (relu) v = fmaxf(v, 0.f);
  Y[t] = v;
}

// ---------------- max pool over neighborhood K ----------------------------------
__global__ void k_maxpool(const float* __restrict__ X, int C, long SB, int K, float* __restrict__ O){
  long t = (long)blockIdx.x*blockDim.x + threadIdx.x;
  if (t >= (long)C*SB) return;
  int c = (int)(t / SB); long s = t % SB;
  const float* p = X + (long)c*SB*K + s*K;
  float m = p[0];
  for (int k = 1; k < K; ++k) m = fmaxf(m, p[k]);
  O[(long)c*SB + s] = m;
}

// ---------------- brute-force kNN (sorted ascending, first-index ties) ----------
template<int KK>
__global__ void k_knn(const float* __restrict__ q, int Sq, const float* __restrict__ r,
                      int Sr, int* __restrict__ idx, float* __restrict__ dout){
  int t = blockIdx.x*blockDim.x + threadIdx.x;
  if (t >= Bb*Sq) return;
  int b = t / Sq;
  float qx = q[t], qy = q[Bb*Sq+t], qz = q[2*Bb*Sq+t];
  const float* X = r + b*Sr; const float* Y = r + Bb*Sr + b*Sr; const float* Z = r + 2*Bb*Sr + b*Sr;
  float d[KK]; int id[KK];
  for (int i = 0; i < KK; ++i){ d[i] = 3.4e38f; id[i] = 0; }
  for (int i = 0; i < Sr; ++i){
    float dd = sqr(X[i]-qx)+sqr(Y[i]-qy)+sqr(Z[i]-qz);
    if (dd < d[KK-1]){
      int p = KK-1;
      while (p > 0 && d[p-1] > dd){ d[p] = d[p-1]; id[p] = id[p-1]; --p; }
      d[p] = dd; id[p] = i;
    }
  }
  for (int i = 0; i < KK; ++i){
    idx[(long)t*KK + i] = id[i];
    if (dout) dout[(long)t*KK + i] = d[i];
  }
}

// ---------------- FlowEmbedding grouping ---------------------------------------
__global__ void k_group_fe(const float* __restrict__ pos1, const float* __restrict__ pos2,
                           const float* __restrict__ f1, const float* __restrict__ f2,
                           int S, int C, const int* __restrict__ idx, int K, float* __restrict__ Xo){
  long cols = (long)Bb*S*K;
  long t = (long)blockIdx.x*blockDim.x + threadIdx.x;
  if (t >= cols) return;
  long sBS = t / K;
  int b = (int)(sBS / S);
  long jg = (long)b*S + idx[t];
  for (int c = 0; c < 3; ++c) Xo[(long)c*cols + t] = pos2[c*(Bb*S)+jg] - pos1[c*(Bb*S)+sBS];
  for (int c = 0; c < C; ++c) Xo[(long)(3+c)*cols + t] = f2[(long)c*(Bb*S)+jg];
  for (int c = 0; c < C; ++c) Xo[(long)(3+C+c)*cols + t] = f1[(long)c*(Bb*S)+sBS];
}

// ---------------- SetUpConv grouping: X = concat(group(f2), pos_diff) -----------
__global__ void k_group_su(const float* __restrict__ pos1, int S1, const float* __restrict__ pos2,
                           int S2, const float* __restrict__ f2, int C2,
                           const int* __restrict__ idx, int K, float* __restrict__ Xo){
  long cols = (long)Bb*S1*K;
  long t = (long)blockIdx.x*blockDim.x + threadIdx.x;
  if (t >= cols) return;
  long sBS = t / K;
  int b = (int)(sBS / S1);
  long jg = (long)b*S2 + idx[t];
  for (int c = 0; c < C2; ++c) Xo[(long)c*cols + t] = f2[(long)c*(Bb*S2)+jg];
  for (int c = 0; c < 3; ++c)  Xo[(long)(C2+c)*cols + t] = pos2[c*(Bb*S2)+jg] - pos1[c*(Bb*S1)+sBS];
}

// ---------------- generic contiguous copy (row concatenation) -------------------
__global__ void k_copy_rows(const float* __restrict__ src, float* __restrict__ dst, long n){
  long t = (long)blockIdx.x*blockDim.x + threadIdx.x;
  if (t < n) dst[t] = src[t];
}

// ---------------- 3-NN inverse-distance interpolation + concat f1 ---------------
__global__ void k_fp_interp(const float* __restrict__ f2, int S2, int C,
                            const int* __restrict__ idx, const float* __restrict__ dist,
                            const float* __restrict__ f1, int C1, int S1, float* __restrict__ Xo){
  long cols = (long)Bb*S1;
  long t = (long)blockIdx.x*blockDim.x + threadIdx.x;
  if (t >= cols) return;
  int b = (int)(t / S1);
  float w0 = 1.f/fmaxf(dist[t*3+0], 1e-10f);
  float w1 = 1.f/fmaxf(dist[t*3+1], 1e-10f);
  float w2 = 1.f/fmaxf(dist[t*3+2], 1e-10f);
  float ws = w0+w1+w2; w0 /= ws; w1 /= ws; w2 /= ws;
  long j0 = (long)b*S2 + idx[t*3+0];
  long j1 = (long)b*S2 + idx[t*3+1];
  long j2 = (long)b*S2 + idx[t*3+2];
  for (int c = 0; c < C; ++c){
    const float* r = f2 + (long)c*(Bb*S2);
    Xo[(long)c*cols + t] = w0*r[j0] + w1*r[j1] + w2*r[j2];
  }
  for (int c = 0; c < C1; ++c) Xo[(long)(C+c)*cols + t] = f1[(long)c*cols + t];
}

// ---------------- final store: [3][B*N] -> [B,3,N] (+bias) ----------------------
__global__ void k_final(const float* __restrict__ Y, const float* __restrict__ bias,
                        float* __restrict__ out){
  int t = blockIdx.x*blockDim.x + threadIdx.x;
  int total = Bb*3*Nn;
  if (t >= total) return;
  int i = t % Nn; int c = (t/Nn)%3; int b = t/(3*Nn);
  out[t] = Y[(long)c*(Bb*Nn) + (long)b*Nn + i] + bias[c];
}

// =========================== host orchestration ================================
struct P { const float* w; const float* g; const float* b; const float* bias; int co, ci; };

static inline int gup(long n, int blk){ return (int)((n + blk - 1) / blk); }

static void gemm(const P& L, const float* X, float* Y, long cols, hipStream_t s){
  int  mTiles  = (L.co + 15)/16;
  int  mGroups = (mTiles + 1)/2;          // MT=2
  long nGroups = (cols + 63)/64;          // NT=4
  long waves = (long)mGroups * nGroups;
  k_gemm_wmma<<<gup(waves, 8), 256, 0, s>>>(L.w, X, L.bias, Y, L.co, L.ci, cols);
}
static void bn_relu(const P& L, float* Y, long cols, float* stats, hipStream_t s){
  k_bn_reduce<<<L.co, 256, 0, s>>>(Y, cols, stats);
  k_bn_apply<<<gup((long)L.co*cols, 256), 256, 0, s>>>(Y, cols, L.co, stats, L.g, L.b, 1);
}
static const float* mlp(const std::vector<P>& Ls, const float* X, long cols,
                        float* bufA, float* bufB, float* stats, hipStream_t s){
  const float* cur = X;
  float* outs[2] = {bufA, bufB};
  int pp = 0;
  for (size_t i = 0; i < Ls.size(); ++i){
    float* Y = outs[pp]; pp ^= 1;
    gemm(Ls[i], cur, Y, cols, s);
    bn_relu(Ls[i], Y, cols, stats, s);
    cur = Y;
  }
  return cur;
}

static void sa_stage(const float* xyz, const float* pts, int n, int D,
                     const std::vector<P>& Ls, int S, float radius, int K,
                     float* new_xyz, float* outF,
                     float* Xbuf, float* bufA, float* bufB, float* stats,
                     int* idxS, int* idxK, hipStream_t s){
  k_fps<<<Bb, 256, n*(int)sizeof(float), s>>>(xyz, n, S, idxS);
  k_gather3<<<gup(Bb*S, 256), 256, 0, s>>>(xyz, n, idxS, S, new_xyz);
  k_ball<<<gup(Bb*S, 256), 256, 0, s>>>(xyz, n, new_xyz, S, radius*radius, K, idxK);
  long cols = (long)Bb*S*K;
  k_group_sa<<<gup(cols, 256), 256, 0, s>>>(xyz, pts, n, D, new_xyz, idxK, S, K, Xbuf);
  const float* f = mlp(Ls, Xbuf, cols, bufA, bufB, stats, s);
  int Co = Ls.back().co;
  k_maxpool<<<gup((long)Co*Bb*S, 256), 256, 0, s>>>(f, Co, (long)Bb*S, K, outF);
}

static void su_stage(const float* pos1, int S1, const float* pos2, int S2,
                     const float* f1, int C1, const float* f2, int C2,
                     const std::vector<P>& l1, const std::vector<P>& l2,
                     float* outF, int* knnIdx, float* Xbuf, float* bufA, float* bufB,
                     float* pooled, float* catbuf, float* stats, hipStream_t s){
  const int K = 8;
  k_knn<8><<<gup(Bb*S1, 256), 256, 0, s>>>(pos1, S1, pos2, S2, knnIdx, nullptr);
  long cols = (long)Bb*S1*K;
  k_group_su<<<gup(cols, 256), 256, 0, s>>>(pos1, S1, pos2, S2, f2, C2, knnIdx, K, Xbuf);
  const float* f; int Cf;
  if (l1.empty()){ f = Xbuf; Cf = C2 + 3; }
  else { f = mlp(l1, Xbuf, cols, bufA, bufB, stats, s); Cf = l1.back().co; }
  long pc = (long)Bb*S1;
  k_maxpool<<<gup((long)Cf*pc, 256), 256, 0, s>>>(f, Cf, pc, K, pooled);
  k_copy_rows<<<gup((long)Cf*pc, 256), 256, 0, s>>>(pooled, catbuf, (long)Cf*pc);
  k_copy_rows<<<gup((long)C1*pc, 256), 256, 0, s>>>(f1, catbuf + (long)Cf*pc, (long)C1*pc);
  const float* r = mlp(l2, catbuf, pc, bufA, bufB, stats, s);
  int Co = l2.back().co;
  k_copy_rows<<<gup((long)Co*pc, 256), 256, 0, s>>>(r, outF, (long)Co*pc);
}

extern "C" void kernel_launch(void* const* d_in, const int* in_sizes, int n_in,
                              void* d_out, int out_size, void* d_ws, size_t ws_size,
                              hipStream_t stream){
  (void)in_sizes; (void)n_in; (void)out_size; (void)ws_size;
  // ---- parse params in setup_inputs() dict order ----
  int p = 4;
  auto nx = [&](){ return (const float*)d_in[p++]; };
  auto mk = [&](int ci, std::initializer_list<int> cos, bool bias)->std::vector<P>{
    std::vector<P> v;
    for (int co : cos){
      P L; L.w = nx(); L.g = nx(); L.b = nx(); L.bias = bias ? nx() : nullptr;
      L.ci = ci; L.co = co; v.push_back(L); ci = co;
    }
    return v;
  };
  const float* pc1 = (const float*)d_in[0];
  const float* pc2 = (const float*)d_in[1];
  const float* fe1 = (const float*)d_in[2];
  const float* fe2 = (const float*)d_in[3];
  auto sa1  = mk(6,   {32,32,64},    false);
  auto sa2  = mk(67,  {64,64,128},   false);
  auto sa3  = mk(131, {128,128,256}, false);
  auto sa4  = mk(259, {256,256,512}, false);
  auto feL  = mk(259, {128,128,128}, false);
  // su1_1 is empty -> consumes no params
  auto su12 = mk(771, {256,256},     false);
  auto su21 = mk(259, {128,128,256}, false);
  auto su22 = mk(512, {256},         false);
  auto su31 = mk(259, {128,128,256}, false);
  auto su32 = mk(320, {256},         false);
  auto fpL  = mk(259, {256,256},     true);
  P head1; head1.w = nx(); head1.g = nx(); head1.b = nx(); head1.bias = nullptr; head1.ci = 256; head1.co = 128;
  const float* head_w2 = nx(); const float* head_bias2 = nx();
  P head2; head2.w = head_w2; head2.g = nullptr; head2.b = nullptr; head2.bias = nullptr; head2.ci = 128; head2.co = 3;

  // ---- workspace bump allocator ----
  char* base = (char*)d_ws; size_t off = 0;
  auto alf = [&](size_t n)->float*{ float* r = (float*)(base+off); off += ((n*4 + 255)/256)*256; return r; };
  auto ali = [&](size_t n)->int*  { int*   r = (int*)  (base+off); off += ((n*4 + 255)/256)*256; return r; };

  float* xyz1 = alf(3*Bb*Nn); float* xyz2 = alf(3*Bb*Nn);
  float* ft1  = alf(3*Bb*Nn); float* ft2  = alf(3*Bb*Nn);
  float* l1pc1 = alf(3*Bb*1024); float* l1f1 = alf((size_t)64*Bb*1024);
  float* l1pc2 = alf(3*Bb*1024); float* l1f2 = alf((size_t)64*Bb*1024);
  float* l2pc1 = alf(3*Bb*256);  float* l2f1 = alf((size_t)128*Bb*256);
  float* l2pc2 = alf(3*Bb*256);  float* l2f2 = alf((size_t)128*Bb*256);
  float* l2f1n = alf((size_t)128*Bb*256);
  float* l2cat = alf((size_t)256*Bb*256);
  float* l3pc1 = alf(3*Bb*64);   float* l3f1 = alf((size_t)256*Bb*64);
  float* l4pc1 = alf(3*Bb*16);   float* l4f1 = alf((size_t)512*Bb*16);
  float* l3fn  = alf((size_t)256*Bb*64);
  float* l2fn  = alf((size_t)256*Bb*256);
  float* l1fn  = alf((size_t)256*Bb*1024);
  float* stats = alf(1024);
  float* y3    = alf((size_t)3*Bb*Nn);
  int* fpsIdx  = ali(Bb*1024);
  int* nbrIdx  = ali((size_t)Bb*1024*16);          // max of ball(1024*16)/fe-knn(256*64)/fp-knn
  float* knnD  = alf((size_t)Bb*Nn*3);
  float* pooled = alf((size_t)256*Bb*1024);        // >= max pooled block
  float* catbuf = alf((size_t)320*Bb*1024);        // >= max concat block
  float* Xbuf  = alf((size_t)259*Bb*256*64);       // 259 x 131072 (fe, the max)
  float* bufA  = alf((size_t)128*Bb*256*64);       // 16.78M floats
  float* bufB  = alf((size_t)128*Bb*256*64);

  // ---- input layout conversion ----
  k_chanmajor<<<gup(Bb*3*Nn,256),256,0,stream>>>(pc1, xyz1, Nn);
  k_chanmajor<<<gup(Bb*3*Nn,256),256,0,stream>>>(pc2, xyz2, Nn);
  k_chanmajor<<<gup(Bb*3*Nn,256),256,0,stream>>>(fe1, ft1, Nn);
  k_chanmajor<<<gup(Bb*3*Nn,256),256,0,stream>>>(fe2, ft2, Nn);

  // ---- SA levels 1-2 for both clouds ----
  sa_stage(xyz1, ft1, Nn,   3,  sa1, 1024, 0.5f, 16, l1pc1, l1f1, Xbuf, bufA, bufB, stats, fpsIdx, nbrIdx, stream);
  sa_stage(l1pc1, l1f1, 1024, 64, sa2, 256, 1.0f, 16, l2pc1, l2f1, Xbuf, bufA, bufB, stats, fpsIdx, nbrIdx, stream);
  sa_stage(xyz2, ft2, Nn,   3,  sa1, 1024, 0.5f, 16, l1pc2, l1f2, Xbuf, bufA, bufB, stats, fpsIdx, nbrIdx, stream);
  sa_stage(l1pc2, l1f2, 1024, 64, sa2, 256, 1.0f, 16, l2pc2, l2f2, Xbuf, bufA, bufB, stats, fpsIdx, nbrIdx, stream);

  // ---- FlowEmbedding (knn=64, concat) ----
  {
    const int S = 256, K = 64, C = 128;
    k_knn<64><<<gup(Bb*S,256),256,0,stream>>>(l2pc1, S, l2pc2, S, nbrIdx, nullptr);
    long cols = (long)Bb*S*K;
    k_group_fe<<<gup(cols,256),256,0,stream>>>(l2pc1, l2pc2, l2f1, l2f2, S, C, nbrIdx, K, Xbuf);
    const float* f = mlp(feL, Xbuf, cols, bufA, bufB, stats, stream);
    k_maxpool<<<gup((long)128*Bb*S,256),256,0,stream>>>(f, 128, (long)Bb*S, K, l2f1n);
  }

  // ---- SA levels 3-4 on cloud 1 ----
  sa_stage(l2pc1, l2f1n, 256, 128, sa3, 64, 2.0f, 8, l3pc1, l3f1, Xbuf, bufA, bufB, stats, fpsIdx, nbrIdx, stream);
  sa_stage(l3pc1, l3f1,  64,  256, sa4, 16, 4.0f, 8, l4pc1, l4f1, Xbuf, bufA, bufB, stats, fpsIdx, nbrIdx, stream);

  // ---- SetUpConv chain ----
  std::vector<P> empty;
  su_stage(l3pc1, 64,  l4pc1, 16,  l3f1, 256, l4f1, 512, empty, su12,
           l3fn, nbrIdx, Xbuf, bufA, bufB, pooled, catbuf, stats, stream);
  {
    long pc = (long)Bb*256;
    k_copy_rows<<<gup((long)128*pc,256),256,0,stream>>>(l2f1,  l2cat,                  (long)128*pc);
    k_copy_rows<<<gup((long)128*pc,256),256,0,stream>>>(l2f1n, l2cat + (long)128*pc,   (long)128*pc);
  }
  su_stage(l2pc1, 256, l3pc1, 64,  l2cat, 256, l3fn, 256, su21, su22,
           l2fn, nbrIdx, Xbuf, bufA, bufB, pooled, catbuf, stats, stream);
  su_stage(l1pc1, 1024, l2pc1, 256, l1f1, 64,  l2fn, 256, su31, su32,
           l1fn, nbrIdx, Xbuf, bufA, bufB, pooled, catbuf, stats, stream);

  // ---- FeaturePropagation: 3-NN interp + concat(feature1) + MLP ----
  {
    k_knn<3><<<gup(Bb*Nn,256),256,0,stream>>>(xyz1, Nn, l1pc1, 1024, nbrIdx, knnD);
    k_fp_interp<<<gup((long)Bb*Nn,256),256,0,stream>>>(l1fn, 1024, 256, nbrIdx, knnD, ft1, 3, Nn, Xbuf);
    const float* l0 = mlp(fpL, Xbuf, (long)Bb*Nn, bufA, bufB, stats, stream);
    // ---- head ----
    gemm(head1, l0, bufA, (long)Bb*Nn, stream);
    bn_relu(head1, bufA, (long)Bb*Nn, stats, stream);
    gemm(head2, bufA, y3, (long)Bb*Nn, stream);
    k_final<<<gup(Bb*3*Nn,256),256,0,stream>>>(y3, head_bias2, (float*)d_out);
  }
}